// RoiPoolingConv_78889959293186
// MI455X (gfx1250) — compile-verified
//
#include <hip/hip_runtime.h>

// ROI pooling (legacy bilinear resize), MI455X / gfx1250.
// img:  [1, 64, 64, 1024] f32
// rois: [1, 300, 4]       f32  (x, y, w, h)
// out:  [1, 300, 14, 14, 1024] f32
//
// Bandwidth-bound gather kernel: image (16.8 MB) is L2-resident; output
// (240 MB) is streamed with non-temporal b128 stores to avoid evicting it.

#define POOL 14
#define HW   64
#define CH   1024
#define NROI 300

typedef float v4f __attribute__((ext_vector_type(4)));

__global__ __launch_bounds__(256) void roi_pool_kernel(
    const float* __restrict__ img,   // [64,64,1024]
    const float* __restrict__ rois,  // [300,4]
    float* __restrict__ out)         // [300,14,14,1024]
{
    const int roi = blockIdx.x / POOL;
    const int py  = blockIdx.x - roi * POOL;

    // ROI params: uniform across block -> scalar loads / SALU math.
    const int rx = (int)rois[roi * 4 + 0];
    const int ry = (int)rois[roi * 4 + 1];
    const int rw = (int)rois[roi * 4 + 2];
    const int rh = (int)rois[roi * 4 + 3];

    // y-direction lerp coords (match reference f32 ops exactly).
    const float sy  = (float)rh / (float)POOL;
    const float fy  = (float)py * sy;
    const int   jy0 = (int)floorf(fy);
    const int   jy1 = min(jy0 + 1, rh - 1);
    const float wy  = fy - (float)jy0;
    const int   y0  = min(max(ry + jy0, 0), HW - 1);
    const int   y1  = min(max(ry + jy1, 0), HW - 1);

    const float sx = (float)rw / (float)POOL;

    const int c = threadIdx.x * 4;  // 256 lanes * 4 ch = 1024 channels

    const float* __restrict__ rowt = img + (size_t)y0 * (HW * CH);
    const float* __restrict__ rowb = img + (size_t)y1 * (HW * CH);
    float* __restrict__ orow = out + ((size_t)roi * POOL + (size_t)py) * (POOL * CH);

    #pragma unroll 2
    for (int px = 0; px < POOL; ++px) {
        const float fx  = (float)px * sx;
        const int   jx0 = (int)floorf(fx);
        const int   jx1 = min(jx0 + 1, rw - 1);
        const float wx  = fx - (float)jx0;
        const int   x0  = min(max(rx + jx0, 0), HW - 1);
        const int   x1  = min(max(rx + jx1, 0), HW - 1);

        // 4 coalesced b128 gathers (L2-resident image).
        const v4f tl = *(const v4f*)(rowt + x0 * CH + c);
        const v4f tr = *(const v4f*)(rowt + x1 * CH + c);
        const v4f bl = *(const v4f*)(rowb + x0 * CH + c);
        const v4f br = *(const v4f*)(rowb + x1 * CH + c);

        // Same op order as reference: top = tl + (tr-tl)*wx; ...
        const v4f top = tl + (tr - tl) * wx;
        const v4f bot = bl + (br - bl) * wx;
        const v4f o   = top + (bot - top) * wy;

        // Streaming write-once output: non-temporal b128 store.
        __builtin_nontemporal_store(o, (v4f*)(orow + px * CH + c));
    }
}

extern "C" void kernel_launch(void* const* d_in, const int* in_sizes, int n_in,
                              void* d_out, int out_size, void* d_ws, size_t ws_size,
                              hipStream_t stream) {
    const float* img  = (const float*)d_in[0];   // [1,64,64,1024] f32
    const float* rois = (const float*)d_in[1];   // [1,300,4]      f32
    float* out = (float*)d_out;                  // [1,300,14,14,1024] f32

    dim3 grid(NROI * POOL);   // 4200 blocks: one (roi, py) row strip each
    dim3 block(256);          // 8 wave32s; lane covers 4 channels (b128)
    roi_pool_kernel<<<grid, block, 0, stream>>>(img, rois, out);
}